// Dynamic_FC_ADMIT_7327214207415
// MI455X (gfx1250) — compile-verified
//
#include <hip/hip_runtime.h>

typedef float v2f __attribute__((ext_vector_type(2)));
typedef float v8f __attribute__((ext_vector_type(8)));

#define IND     64
#define OUTD    64
#define NBASIS  5
#define XSTRIDE 65       // x and out row stride in floats (t + 64 features/outputs)
#define LSTRIDE 642      // floats per lane stripe in LDS (640 data + 2 pad -> bank spread)

// Each wave computes a full 16-row x 64-col output tile with
// V_WMMA_F32_16X16X4_F32, folding the 5 basis functions into scaled A operands.
// Weight is pre-permuted into a lane-major LDS layout so every B fragment is a
// ds_load_b64 from one per-lane base + 16-bit immediate (no address ALU).
__global__ __launch_bounds__(256) void spline_fc_wmma(
    const float* __restrict__ x,
    const float* __restrict__ weight,
    const float* __restrict__ bias,
    float* __restrict__ out,
    int nblocks)
{
  __shared__ float Wl[32 * LSTRIDE];   // 82,176 B

  const int tid = threadIdx.x;

  // One-time permutation: Wl[l][((d*4+nt)*16+j)*2 + p] = weight[4j+2*hi(l)+p, nt*16+n(l), d]
  for (int idx = tid; idx < 32 * 640; idx += blockDim.x) {
    int l  = idx / 640;
    int r  = idx - l * 640;
    int p  = r & 1;
    int pr = r >> 1;               // (d*4+nt)*16 + j
    int d  = pr >> 6;
    int r2 = pr & 63;
    int nt = r2 >> 4;
    int j  = r2 & 15;
    int hi = l >> 4;
    int n  = l & 15;
    int i  = 4 * j + 2 * hi + p;   // K index
    Wl[l * LSTRIDE + r] = weight[i * (OUTD * NBASIS) + (nt * 16 + n) * NBASIS + d];
  }
  __syncthreads();

  const int lane   = tid & 31;
  const int waveid = tid >> 5;
  const int n      = lane & 15;   // A: row within tile / B,C: column within n-tile
  const int hi     = lane >> 4;   // lane half
  const int kofs   = hi * 2;      // fp32 WMMA K offset per lane half

  // Hoist bias fragments: biasv[nt][d] = bias[nt*16+n][d]
  float biasv[4][NBASIS];
#pragma unroll
  for (int nt = 0; nt < 4; ++nt)
#pragma unroll
    for (int d = 0; d < NBASIS; ++d)
      biasv[nt][d] = bias[(nt * 16 + n) * NBASIS + d];

  const int gw      = blockIdx.x * 8 + waveid;
  const int wstride = gridDim.x * 8;

  for (int rb = gw; rb < nblocks; rb += wstride) {
    const int rowbase = rb * 16;
    const int m       = rowbase + n;
    const float* xrow = x + (long)m * XSTRIDE;

    // Prefetch next row-block this wave will touch
    __builtin_prefetch(x + (long)(m + wstride * 16) * XSTRIDE, 0, 0);

    // Opaque zero: keeps the 320 B-fragment ds_loads inside the loop (blocks
    // LICM -> scratch spilling). Scaled by 2 so the LDS index stays provably
    // even and the compiler can emit aligned ds_load_b64.
    int licm = 0;
    asm volatile("" : "+v"(licm));
    const float* wlane = &Wl[lane * LSTRIDE + 2 * licm];

    // A fragments: a[j] = (x_feat[m, 4j+kofs], x_feat[m, 4j+kofs+1])
    v2f a[16];
#pragma unroll
    for (int j = 0; j < 16; ++j) {
      a[j].x = xrow[1 + 4 * j + kofs];
      a[j].y = xrow[1 + 4 * j + kofs + 1];
    }

    // Basis scales for this lane's A row (row = rowbase + n)
    const float tm = xrow[0];
    float s[NBASIS];
    s[0] = 1.0f;
    s[1] = tm;
    s[2] = tm * tm;
    float r1 = fmaxf(tm - 0.33f, 0.0f); s[3] = r1 * r1;
    float r2 = fmaxf(tm - 0.66f, 0.0f); s[4] = r2 * r2;

    v8f C[4] = {};   // 16x64 output tile, 4 n-tiles

#pragma unroll
    for (int d = 0; d < NBASIS; ++d) {
#pragma unroll
      for (int j = 0; j < 16; ++j) {
        v2f aa = a[j] * s[d];                       // fold basis into A
#pragma unroll
        for (int nt = 0; nt < 4; ++nt) {
          // single base + 16-bit immediate: ds_load_b64
          v2f bb = *(const v2f*)&wlane[((d * 4 + nt) * 16 + j) * 2];
          C[nt] = __builtin_amdgcn_wmma_f32_16x16x4_f32(
              false, aa, false, bb, (short)0, C[nt], false, false);
        }
      }
    }

    // Epilogue: + basis @ bias.T, relu, prepend t.
    // t for C-row (v + 8*hi) lives in lane (v + 8*hi): shuffle, don't reload.
#pragma unroll
    for (int v = 0; v < 8; ++v) {
      const int row  = rowbase + v + 8 * hi;
      const float tr = __shfl(tm, v + 8 * hi, 32);
      const float b2 = tr * tr;
      float q1 = fmaxf(tr - 0.33f, 0.0f); const float b3 = q1 * q1;
      float q2 = fmaxf(tr - 0.66f, 0.0f); const float b4 = q2 * q2;
      float* orow = out + (long)row * XSTRIDE;
      if (n == 0) orow[0] = tr;
#pragma unroll
      for (int nt = 0; nt < 4; ++nt) {
        float acc = C[nt][v]
                  + biasv[nt][0] + biasv[nt][1] * tr + biasv[nt][2] * b2
                  + biasv[nt][3] * b3 + biasv[nt][4] * b4;
        orow[1 + nt * 16 + n] = fmaxf(acc, 0.0f);
      }
    }
  }
}

// Scalar fallback for a non-multiple-of-16 row tail (not hit for B=500000).
__global__ void spline_fc_tail(
    const float* __restrict__ x,
    const float* __restrict__ weight,
    const float* __restrict__ bias,
    float* __restrict__ out,
    int rowstart, int rows)
{
  int t_ = blockIdx.x * blockDim.x + threadIdx.x;
  if (t_ >= rows * OUTD) return;
  int row = rowstart + t_ / OUTD;
  int o   = t_ - (t_ / OUTD) * OUTD;
  const float* xr = x + (long)row * XSTRIDE;
  float tv = xr[0];
  float s[NBASIS];
  s[0] = 1.0f; s[1] = tv; s[2] = tv * tv;
  float r1 = fmaxf(tv - 0.33f, 0.0f); s[3] = r1 * r1;
  float r2 = fmaxf(tv - 0.66f, 0.0f); s[4] = r2 * r2;
  float acc = 0.0f;
  for (int d = 0; d < NBASIS; ++d) acc += s[d] * bias[o * NBASIS + d];
  for (int i = 0; i < IND; ++i) {
    float xf = xr[1 + i];
    for (int d = 0; d < NBASIS; ++d)
      acc += xf * weight[(i * OUTD + o) * NBASIS + d] * s[d];
  }
  float* orow = out + (long)row * XSTRIDE;
  if (o == 0) orow[0] = tv;
  orow[1 + o] = fmaxf(acc, 0.0f);
}

extern "C" void kernel_launch(void* const* d_in, const int* in_sizes, int n_in,
                              void* d_out, int out_size, void* d_ws, size_t ws_size,
                              hipStream_t stream) {
  const float* x      = (const float*)d_in[0];   // (B, 65)
  const float* weight = (const float*)d_in[1];   // (64, 64, 5)
  const float* bias   = (const float*)d_in[2];   // (64, 5)
  float* out          = (float*)d_out;           // (B, 65)

  const int B       = in_sizes[0] / XSTRIDE;
  const int nblocks = B / 16;
  const int tail    = B - nblocks * 16;

  if (nblocks > 0) {
    dim3 grid(1024), block(256);
    hipLaunchKernelGGL(spline_fc_wmma, grid, block, 0, stream,
                       x, weight, bias, out, nblocks);
  }
  if (tail > 0) {
    int work = tail * OUTD;
    hipLaunchKernelGGL(spline_fc_tail, dim3((work + 255) / 256), dim3(256), 0, stream,
                       x, weight, bias, out, nblocks * 16, tail);
  }
}